// NDDE_1D_47064251630184
// MI455X (gfx1250) — compile-verified
//
#include <hip/hip_runtime.h>
#include <hip/hip_bf16.h>

// ---------------------------------------------------------------------------
// NDDE forward-Euler solver for MI455X (gfx1250).
//
// Strategy (see analysis): weights resident in LDS (fp32, WMMA-A-swizzled),
// 48 WGs x 16 rows, K=1536 split over 8 waves, V_WMMA_F32_16X16X4_F32 with
// the state vector broadcast across B's 16 columns. Per-step cross-WG
// exchange of the 3KB state via an L2 ring buffer + per-step atomic barrier.
// ---------------------------------------------------------------------------

#define D_DIM   768
#define NSTEPS  8192
#define NP1     (NSTEPS + 1)
#define GWG     48                      // workgroups (768 / 16 rows)
#define ROWS    16                      // output rows per WG (one WMMA M-block)
#define NT      256                     // threads per WG = 8 wave32
#define NWAVE   8
#define KW      192                     // concat-K handled per wave (8*192 = 1536)
#define NPAIR   24                      // WMMA tile-pairs per wave (pair = 2 x K=4)
#define RS      16                      // ring slots (must be > 10 + max skew)
#define W_FLOATS (NWAVE * ROWS * KW)    // 24576 floats = 96 KB of swizzled weights
#define SMEM_FLOATS (W_FLOATS + 2 * D_DIM + NWAVE * ROWS)
#define SMEM_BYTES  (SMEM_FLOATS * 4)   // 104,960 B  (< 320 KB WGP LDS)

typedef float v2f __attribute__((ext_vector_type(2)));
typedef float v8f __attribute__((ext_vector_type(8)));

__global__ void ndde_zero_cnt(unsigned* __restrict__ cnt) {
    int i = blockIdx.x * blockDim.x + threadIdx.x;
    if (i < NSTEPS) cnt[i] = 0u;
}

__global__ __launch_bounds__(NT, 1)
void ndde_solve(const float* __restrict__ x0, const float* __restrict__ tau,
                const float* __restrict__ Wx, const float* __restrict__ Wy,
                const float* __restrict__ bb, float* __restrict__ out,
                unsigned* __restrict__ cnt, float* __restrict__ ring) {
    extern __shared__ float smem[];
    float* wsw  = smem;                  // [NWAVE][NPAIR][32 lanes][4] swizzled W
    float* xbuf = smem + W_FLOATS;       // current state x_j        (768)
    float* ybuf = xbuf + D_DIM;          // delayed state x_{j-10}   (768)
    float* zred = ybuf + D_DIM;          // per-wave partial z [NWAVE][16]

    const int g    = blockIdx.x;
    const int tid  = threadIdx.x;
    const int w    = tid >> 5;           // wave id 0..7
    const int lane = tid & 31;
    const int h    = lane >> 4;          // WMMA half: lanes 16-31 hold K+2

    const float dt = tau[0] * 0.1f;                       // dt = tau / N_tau
    const float bm = (tid < ROWS) ? bb[g * ROWS + tid] : 0.f;

    // ---- prologue: init state buffers, ring slot 0, trajectory column 0 ----
    for (int i = tid; i < D_DIM; i += NT) {
        float v = x0[i];
        xbuf[i] = v;
        ybuf[i] = v;                     // history x(t)=x_0 for t<=0
    }
    if (tid < ROWS) {
        int M = g * ROWS + tid;
        float v = x0[M];
        ring[M] = v;                                       // ring slot 0 = col 0
        __builtin_nontemporal_store(v, &out[(size_t)M * NP1]); // traj col 0
    }

    // ---- one-time: swizzle [Wx | Wy] rows into WMMA A-operand LDS layout ----
    // Lane l of tile-pair p stores {W[m][k0+2h], W[m][k0+2h+1],
    //                               W[m][k0+4+2h], W[m][k0+4+2h+1]}
    // so one b128 LDS read feeds the A regs of two 16x16x4 f32 WMMAs.
    for (int i = tid; i < W_FLOATS; i += NT) {
        int wv = i / (ROWS * KW);
        int r  = i - wv * (ROWS * KW);
        int p  = r >> 7;
        int q  = r & 127;
        int l  = q >> 2;
        int jj = q & 3;
        int hh = l >> 4;
        int m  = l & 15;
        int kloc = p * 8 + 2 * hh + ((jj >> 1) << 2) + (jj & 1);
        int kcat = wv * KW + kloc;                         // 0..1535 over [Wx|Wy]
        int M = g * ROWS + m;
        float v = (kcat < D_DIM) ? Wx[(size_t)M * D_DIM + kcat]
                                 : Wy[(size_t)M * D_DIM + (kcat - D_DIM)];
        wsw[i] = v;
    }
    __syncthreads();

    const float4* wp = reinterpret_cast<const float4*>(wsw + w * (ROWS * KW));
    const float*  vb = (w < 4) ? xbuf : ybuf;             // waves 0-3: x, 4-7: y
    const int     kb = (w & 3) * KW;

    for (int j = 0; j < NSTEPS; ++j) {
        // ---- z-block = W_rows @ [x;y] via WMMA, B broadcast across N ----
        v8f a0c = {0.f,0.f,0.f,0.f,0.f,0.f,0.f,0.f};
        v8f a1c = {0.f,0.f,0.f,0.f,0.f,0.f,0.f,0.f};
        v8f a2c = {0.f,0.f,0.f,0.f,0.f,0.f,0.f,0.f};
        v8f a3c = {0.f,0.f,0.f,0.f,0.f,0.f,0.f,0.f};
        #pragma unroll
        for (int p = 0; p < NPAIR; p += 2) {
            float4 wa = wp[p * 32 + lane];
            int k0 = kb + p * 8 + 2 * h;
            v2f a0 = {wa.x, wa.y};
            v2f b0 = {vb[k0], vb[k0 + 1]};
            v2f a1 = {wa.z, wa.w};
            v2f b1 = {vb[k0 + 4], vb[k0 + 5]};
            a0c = __builtin_amdgcn_wmma_f32_16x16x4_f32(false, a0, false, b0,
                                                        (short)0, a0c, false, false);
            a1c = __builtin_amdgcn_wmma_f32_16x16x4_f32(false, a1, false, b1,
                                                        (short)0, a1c, false, false);
            float4 wb = wp[(p + 1) * 32 + lane];
            int k1 = kb + (p + 1) * 8 + 2 * h;
            v2f a2 = {wb.x, wb.y};
            v2f b2 = {vb[k1], vb[k1 + 1]};
            v2f a3 = {wb.z, wb.w};
            v2f b3 = {vb[k1 + 4], vb[k1 + 5]};
            a2c = __builtin_amdgcn_wmma_f32_16x16x4_f32(false, a2, false, b2,
                                                        (short)0, a2c, false, false);
            a3c = __builtin_amdgcn_wmma_f32_16x16x4_f32(false, a3, false, b3,
                                                        (short)0, a3c, false, false);
        }
        v8f d = a0c + a1c + a2c + a3c;

        // C/D f32 16x16 layout: lane 0 VGPR r -> z[r], lane 16 VGPR r -> z[r+8]
        // (every N column identical since B was broadcast).
        if (lane == 0) {
            #pragma unroll
            for (int r = 0; r < 8; ++r) zred[w * 16 + r] = d[r];
        } else if (lane == 16) {
            #pragma unroll
            for (int r = 0; r < 8; ++r) zred[w * 16 + 8 + r] = d[r];
        }
        __syncthreads();

        // ---- reduce 8 wave-partials, Euler update, publish row chunk ----
        if (tid < ROWS) {
            float z = bm;
            #pragma unroll
            for (int w2 = 0; w2 < NWAVE; ++w2) z += zred[w2 * 16 + tid];
            int M = g * ROWS + tid;
            float xn = xbuf[M] + dt * tanhf(z);
            ring[(size_t)((j + 1) & (RS - 1)) * D_DIM + M] = xn;        // ring col j+1
            __builtin_nontemporal_store(xn, &out[(size_t)M * NP1 + (j + 1)]);
        }

        __threadfence();             // release our ring/out stores device-wide
        __syncthreads();
        if (j == NSTEPS - 1) break;  // last column written; no more sync needed

        // ---- grid barrier for step j (one leader per WG) ----
        if (tid == 0) {
            __hip_atomic_fetch_add(&cnt[j], 1u, __ATOMIC_RELEASE,
                                   __HIP_MEMORY_SCOPE_AGENT);
            while (__hip_atomic_load(&cnt[j], __ATOMIC_ACQUIRE,
                                     __HIP_MEMORY_SCOPE_AGENT) < (unsigned)GWG) {
                __builtin_amdgcn_s_sleep(2);
            }
        }
        __syncthreads();
        __threadfence();             // acquire: invalidate stale state lines

        // ---- fetch full x_{j+1} and delayed y_{j+1} from the ring ----
        int jc = j + 1;
        const float* rx = ring + (size_t)(jc & (RS - 1)) * D_DIM;
        int ycol = (jc >= 10) ? (jc - 10) : 0;             // y = x(t - tau)
        const float* ry = ring + (size_t)(ycol & (RS - 1)) * D_DIM;
        for (int i = tid; i < D_DIM; i += NT) {
            xbuf[i] = rx[i];
            ybuf[i] = ry[i];
        }
        __syncthreads();
    }
}

extern "C" void kernel_launch(void* const* d_in, const int* in_sizes, int n_in,
                              void* d_out, int out_size, void* d_ws, size_t ws_size,
                              hipStream_t stream) {
    const float* x0  = (const float*)d_in[0];
    const float* tau = (const float*)d_in[1];
    const float* Wx  = (const float*)d_in[2];
    const float* Wy  = (const float*)d_in[3];
    const float* bb  = (const float*)d_in[4];
    float* out = (float*)d_out;

    // ws layout: [0, 32768) per-step counters, [32768, 32768+49152) state ring.
    unsigned* cnt = (unsigned*)d_ws;
    float* ring   = (float*)((char*)d_ws + (size_t)NSTEPS * sizeof(unsigned));

    // 104,960 B dynamic LDS per WG (> default 64KB cap) — raise the limit.
    (void)hipFuncSetAttribute(reinterpret_cast<const void*>(&ndde_solve),
                              hipFuncAttributeMaxDynamicSharedMemorySize,
                              SMEM_BYTES);

    ndde_zero_cnt<<<(NSTEPS + 255) / 256, 256, 0, stream>>>(cnt);
    ndde_solve<<<GWG, NT, SMEM_BYTES, stream>>>(x0, tau, Wx, Wy, bb, out, cnt, ring);
}